// FeedForwardNetwork_81363860455912
// MI455X (gfx1250) — compile-verified
//
#include <hip/hip_runtime.h>

#define B_SZ   1024
#define N_INP  256
#define LAYERS 8
#define M_N    4096
#define K_N    16
#define SCALE_F 4.9f
// Columns kept in the workspace table: input(256)+bias(1)+7 hidden layers.
// Layer 7's output goes only to d_out (nothing reads it afterwards).
#define NCOLS (N_INP + 1 + (LAYERS - 1) * M_N)   // 28929  (~113 MB table, fits L2)

typedef float v2f __attribute__((ext_vector_type(2)));
typedef float v8f __attribute__((ext_vector_type(8)));

// Build the column-major activation table: T[c][b], c in [0,257).
__global__ void init_table(const float* __restrict__ x, float* __restrict__ T) {
    int c = blockIdx.x;                                   // 0..256
    int b = blockIdx.y * blockDim.x + threadIdx.x;        // 0..1023
    float v = (c == N_INP) ? 1.0f : x[(size_t)b * N_INP + c];
    T[(size_t)c * B_SZ + b] = v;
}

// sigmoid(SCALE*z) = 1 / (1 + exp2(z * (-SCALE*log2(e)))) : 1 mul + 1 v_exp +
// 1 add + 1 v_rcp (fast HW ops, ~1 ulp).
__device__ __forceinline__ float fast_sigmoid_scaled(float z) {
    const float C = -4.9f * 1.44269504088896340736f;     // folded at compile time
    float e = __builtin_amdgcn_exp2f(z * C);
    return __builtin_amdgcn_rcpf(1.0f + e);
}

// One layer: pre[b,m] = sum_k W[m,k] * T[idx[m,k]][b]; out = sigmoid(SCALE*pre).
// 16x16 (batch x neuron) tiles via exact-f32 V_WMMA_F32_16X16X4_F32 with
// single-column B fragments built from pre-hoisted per-lane weight registers.
__global__ __launch_bounds__(128) void ffn_layer(
    float* T,                         // column-major table (read < colbase, write >= colbase)
    const float* __restrict__ Wl,     // W[l]   : [M][K] row-major
    const int*   __restrict__ idxl,   // idx[l] : [M][K] row-major
    float* __restrict__ outp,         // d_out (row-major B x M) when last
    int colbase, int last) {
    __shared__ int   sIdx[256];       // 16 neurons x 16 k
    __shared__ float sW[256];
    __shared__ float sT[4][16 * 17];  // per-wave transpose tile, padded

    const int tid  = threadIdx.x;
    const int lane = tid & 31;
    const int wave = tid >> 5;
    const int hi   = lane >> 4;       // lane half: 0 or 1
    const int lm   = lane & 15;
    const int mbase = blockIdx.x * 16;

    // Stage this m-tile's indices & weights once per block (coalesced).
    for (int t = tid; t < 256; t += 128) {
        sIdx[t] = idxl[mbase * K_N + t];
        sW[t]   = Wl[mbase * K_N + t];
    }
    __syncthreads();

    // Per-lane B-fragment weights, hoisted out of the batch loop:
    // for chunk j, lane (hi,lm) supplies rows K=4j+2*hi, 4j+2*hi+1 of column lm:
    //   wfrag[j] = { W[lm][4j+2*hi], W[lm][4j+2*hi+1] }
    v2f wfrag[4];
    #pragma unroll
    for (int j = 0; j < 4; ++j) {
        wfrag[j].x = sW[lm * 16 + 4 * j + 2 * hi + 0];
        wfrag[j].y = sW[lm * 16 + 4 * j + 2 * hi + 1];
    }

    const char* Tb = (const char*)T;  // 32-bit byte offsets off one SGPR base

    for (int it = 0; it < 4; ++it) {
        const int btile = blockIdx.y * 16 + it * 4 + wave;   // 0..63
        const int b0 = btile * 16;
        const unsigned boff = (unsigned)(b0 + lm) * 4u;

        v8f acc = {};
        #pragma unroll 2
        for (int n = 0; n < 16; ++n) {
            const bool isn = (lm == n);        // one v_cmp per n
            #pragma unroll
            for (int j = 0; j < 4; ++j) {
                // A (16x4 f32): lanes0-15 rows use K=4j+0/1, lanes16-31 K=4j+2/3.
                // Per-lane idx pair: one ds_load_b64 (half-wave broadcast).
                int2 cc = *(const int2*)&sIdx[n * 16 + 4 * j + 2 * hi];
                v2f a;
                a.x = *(const float*)(Tb + ((unsigned)cc.x * (B_SZ * 4u) + boff));
                a.y = *(const float*)(Tb + ((unsigned)cc.y * (B_SZ * 4u) + boff));
                // B (4x16 f32): only column n nonzero; pure register selects.
                v2f bf;
                bf.x = isn ? wfrag[j].x : 0.0f;
                bf.y = isn ? wfrag[j].y : 0.0f;
                acc = __builtin_amdgcn_wmma_f32_16x16x4_f32(
                    false, a, false, bf, (short)0, acc, false, false);
            }
        }

        // D layout: VGPR r -> (b_local = hi*8 + r, n = lm). Apply sigmoid.
        float sg[8];
        #pragma unroll
        for (int r = 0; r < 8; ++r)
            sg[r] = fast_sigmoid_scaled(acc[r]);

        if (last) {
            // d_out row-major [b][m]: contiguous across each lane half.
            #pragma unroll
            for (int r = 0; r < 8; ++r) {
                int b = b0 + hi * 8 + r;
                outp[(size_t)b * M_N + mbase + lm] = sg[r];
            }
        } else {
            // Transpose through LDS so column-major stores are coalesced.
            #pragma unroll
            for (int r = 0; r < 8; ++r)
                sT[wave][lm * 17 + hi * 8 + r] = sg[r];
            __syncthreads();
            #pragma unroll
            for (int p = 0; p < 8; ++p) {
                int n = 2 * p + hi;
                float v = sT[wave][n * 17 + lm];
                T[(size_t)(colbase + mbase + n) * B_SZ + b0 + lm] = v;
            }
            __syncthreads();
        }
    }
}

extern "C" void kernel_launch(void* const* d_in, const int* in_sizes, int n_in,
                              void* d_out, int out_size, void* d_ws, size_t ws_size,
                              hipStream_t stream) {
    (void)in_sizes; (void)n_in; (void)out_size;
    const float* x   = (const float*)d_in[0];   // (B, 256) f32
    const float* W   = (const float*)d_in[1];   // (L, M, K) f32
    const int*   idx = (const int*)d_in[2];     // (L, M, K) i32
    float* out = (float*)d_out;                 // (B, M) f32
    float* T   = (float*)d_ws;                  // column-major table [NCOLS][B]

    size_t need = (size_t)NCOLS * B_SZ * sizeof(float);
    if (ws_size < need) return;                 // cannot run without table storage

    init_table<<<dim3(N_INP + 1, B_SZ / 256), 256, 0, stream>>>(x, T);

    for (int l = 0; l < LAYERS; ++l) {
        int colbase = N_INP + 1 + l * M_N;
        int last = (l == LAYERS - 1);
        ffn_layer<<<dim3(M_N / 16, 4), 128, 0, stream>>>(
            T,
            W   + (size_t)l * M_N * K_N,
            idx + (size_t)l * M_N * K_N,
            last ? out : nullptr,
            colbase, last);
    }
}